// CrowdInteraction_9320079033050
// MI455X (gfx1250) — compile-verified
//
#include <hip/hip_runtime.h>

typedef float v2f __attribute__((ext_vector_type(2)));
typedef float v4f __attribute__((ext_vector_type(4)));
typedef float v8f __attribute__((ext_vector_type(8)));

#define PED_N 8192
#define HID   64
#define MT    32                 // rows of C per wave (2 x 16-row WMMA blocks)
#define SPLITK 16                // waves per block, each owns a K chunk of 512
#define KC    (PED_N / SPLITK)   // 512
#define TILE_ELEMS (MT * HID)    // 2048 floats = 8 KB per wave partial

// Block = 512 threads = 16 waves. blockIdx.x selects the 32-row tile of C;
// wave w computes the partial product over K in [w*512, (w+1)*512), stores its
// 32x64 partial into a private LDS region, then the block sums the 16
// partials deterministically and writes C with wide coalesced stores.
__global__ void __launch_bounds__(512)
crowd_gemm_wmma_f32(const float* __restrict__ A,   // [8192, 8192] row-major
                    const float* __restrict__ B,   // [8192, 64]   row-major
                    float* __restrict__ C) {       // [8192, 64]   row-major
    __shared__ float smem[SPLITK * TILE_ELEMS];    // 16 x 8 KB = 128 KB

    const int tid  = threadIdx.x;
    const int wave = tid >> 5;        // 0..15  == K-split index
    const int lane = tid & 31;
    const int lo   = lane & 15;       // 0..15
    const int hi   = lane >> 4;       // 0 or 1

    const int rowBase = blockIdx.x * MT;
    const int kStart  = wave * KC;

    // Accumulators: 2 row-blocks x 4 col-blocks of 16x16 f32
    v8f acc[2][4];
#pragma unroll
    for (int rb = 0; rb < 2; ++rb)
#pragma unroll
        for (int cb = 0; cb < 4; ++cb)
            acc[rb][cb] = (v8f)0.0f;

    // A fragment (16x4 f32 ISA layout): lane<16 holds K=kb+0,kb+1 of row M=lane;
    // lane>=16 holds K=kb+2,kb+3. One aligned b64 load per lane per K-step.
    const float* Arow[2];
#pragma unroll
    for (int rb = 0; rb < 2; ++rb)
        Arow[rb] = A + (size_t)(rowBase + rb * 16 + lo) * PED_N + 2 * hi;

    // B fragment (4x16 f32, mirrored layout): v0 = rows kb+0 / kb+2,
    // v1 = rows kb+1 / kb+3, column = cb*16 + lo.
    const float* Bbase = B + (size_t)(2 * hi) * HID + lo;

    // Outer step = 32 K values (one 128B cacheline per A row per lane group).
    for (int kb0 = kStart; kb0 < kStart + KC; kb0 += 32) {
        // Prefetch the A stream 4 KB (1024 K values) ahead into L2.
        // Speculative: silently dropped if it runs past the mapping.
#pragma unroll
        for (int rb = 0; rb < 2; ++rb)
            __builtin_prefetch(Arow[rb] + kb0 + 1024, 0, 1);

#pragma unroll
        for (int j = 0; j < 8; ++j) {
            const int kb = kb0 + 4 * j;

            v2f a[2];
#pragma unroll
            for (int rb = 0; rb < 2; ++rb)
                a[rb] = __builtin_nontemporal_load((const v2f*)(Arow[rb] + kb));

            v2f b[4];
#pragma unroll
            for (int cb = 0; cb < 4; ++cb) {
                const float* p = Bbase + (size_t)kb * HID + cb * 16;
                b[cb].x = p[0];
                b[cb].y = p[HID];
            }

#pragma unroll
            for (int rb = 0; rb < 2; ++rb)
#pragma unroll
                for (int cb = 0; cb < 4; ++cb)
                    acc[rb][cb] = __builtin_amdgcn_wmma_f32_16x16x4_f32(
                        false, a[rb], false, b[cb],
                        (short)0, acc[rb][cb], false, false);
        }
    }

    // ---- Deterministic split-K reduction through LDS (no atomics) ----
    // C/D 16x16 f32 layout: VGPR r -> M = r + 8*hi, lane -> N = lo.
    float* myTile = smem + wave * TILE_ELEMS;
#pragma unroll
    for (int rb = 0; rb < 2; ++rb)
#pragma unroll
        for (int cb = 0; cb < 4; ++cb)
#pragma unroll
            for (int r = 0; r < 8; ++r)
                myTile[(rb * 16 + hi * 8 + r) * HID + cb * 16 + lo] =
                    acc[rb][cb][r];

    __syncthreads();

    // 512 threads x 4 consecutive floats each cover the 2048-float tile.
    // Fixed summation order over the 16 partials -> bitwise deterministic.
    v4f s = (v4f)0.0f;
    const int base = tid * 4;
#pragma unroll
    for (int w = 0; w < SPLITK; ++w)
        s += *(const v4f*)(smem + w * TILE_ELEMS + base);

    // Tile is 32 full rows of C -> contiguous 2048 floats at rowBase*64.
    *(v4f*)(C + (size_t)rowBase * HID + base) = s;
}

extern "C" void kernel_launch(void* const* d_in, const int* in_sizes, int n_in,
                              void* d_out, int out_size, void* d_ws, size_t ws_size,
                              hipStream_t stream) {
    const float* A = (const float*)d_in[0];   // location_data [8192, 8192]
    const float* B = (const float*)d_in[1];   // motion_data   [8192, 64]
    float* C = (float*)d_out;                 // [8192, 64]

    // 256 row tiles of 32 rows; every output element written exactly once.
    crowd_gemm_wmma_f32<<<PED_N / MT, 512, 0, stream>>>(A, B, C);
}